// DigitCaps_39840116638202
// MI455X (gfx1250) — compile-verified
//
#include <hip/hip_runtime.h>
#include <math.h>

// Problem constants (from reference)
#define BB 256
#define RR 1152
#define CC 10
#define OO 16
#define II 8

typedef float v2f __attribute__((ext_vector_type(2)));
typedef float v8f __attribute__((ext_vector_type(8)));
typedef int v4i_async __attribute__((vector_size(4 * sizeof(int))));

#if defined(__AMDGCN__) && __has_builtin(__builtin_amdgcn_global_load_async_to_lds_b128)
#define HAVE_ASYNC_LDS 1
#else
#define HAVE_ASYNC_LDS 0
#endif

#if defined(__AMDGCN__) && __has_builtin(__builtin_amdgcn_s_wait_asynccnt)
#define WAIT_ASYNCCNT(n) __builtin_amdgcn_s_wait_asynccnt(n)
#elif defined(__AMDGCN__)
#define WAIT_ASYNCCNT(n) asm volatile("s_wait_asynccnt %0" ::"n"(n) : "memory")
#else
#define WAIT_ASYNCCNT(n)
#endif

#if HAVE_ASYNC_LDS
// One async instruction: each lane copies 16B global -> LDS (ASYNCcnt-tracked)
// Builtin signature (from hipcc diagnostic): params are typed pointers to a
// 16-byte int vector (global side in addrspace(1), LDS side in addrspace(3)).
__device__ __forceinline__ void async_cp16(const float* g, float* l) {
  __builtin_amdgcn_global_load_async_to_lds_b128(
      (__attribute__((address_space(1))) v4i_async*)(g),
      (__attribute__((address_space(3))) v4i_async*)(l), 0, 0);
}
#endif

#define RCH 8                 // routes per staged chunk
#define XCH (16 * RCH * II)   // 1024 floats: x tile (16 batches x 8 routes x 8)
#define WCH (RCH * OO * II)   // 1024 floats: W tile (8 routes x 16 o x 8 i)

// ---------------------------------------------------------------------------
// s[b,c,o] = sum_r c_ij[r,c] * sum_i W[r,c,o,i] * x[b,r,i]
// One wave computes a 16(batch) x 16(o) tile for one capsule c, looping all R
// routes with two V_WMMA_F32_16X16X4_F32 steps per route (K = I = 8).
// Async path: double-buffered global_load_async_to_lds staging of the x / W
// chunks, synced with s_wait_asynccnt, so copies of chunk k+1 overlap WMMA on
// chunk k. Two accumulators break the WMMA D-chain for XDL pipelining.
// ---------------------------------------------------------------------------
__global__ __launch_bounds__(128) void caps_s_kernel(
    const float* __restrict__ x, const float* __restrict__ W,
    const float* __restrict__ cij, float* __restrict__ s) {
  const int lane = threadIdx.x & 31;
  const int wid  = threadIdx.x >> 5;
  const int c    = blockIdx.x;                 // capsule 0..9
  const int b0   = (blockIdx.y * 4 + wid) * 16;
  const int m = lane & 15;   // M (batch) for A / N (o) for B
  const int h = lane >> 4;   // lane-half selects K pair {0,1} vs {2,3}

  v8f acc0 = {}, acc1 = {};

#if HAVE_ASYNC_LDS
  __shared__ float smem[4 * 2 * (XCH + WCH)];  // 64 KB / block, per-wave slices
  float* xs0 = smem + wid * 2 * (XCH + WCH);
  float* ws0 = xs0 + 2 * XCH;

  auto issue = [&](int r0, int buf) {
    float* xd = xs0 + buf * XCH;
    float* wd = ws0 + buf * WCH;
#pragma unroll
    for (int k = 0; k < 8; ++k) {  // x: two 256B batch-rows per instruction
      const int b = k * 2 + h;
      async_cp16(x + ((size_t)(b0 + b) * RR + r0) * II + m * 4,
                 xd + b * (RCH * II) + m * 4);
    }
#pragma unroll
    for (int k = 0; k < RCH; ++k) {  // W: one 512B route per instruction
      async_cp16(W + ((size_t)(r0 + k) * CC + c) * (OO * II) + lane * 4,
                 wd + k * (OO * II) + lane * 4);
    }
  };

  issue(0, 0);
  const int NCH = RR / RCH;  // 144 chunks
  for (int ch = 0; ch < NCH; ++ch) {
    if (ch + 1 < NCH) {
      issue((ch + 1) * RCH, (ch + 1) & 1);
      WAIT_ASYNCCNT(16);  // chunk ch's 16 copies complete; ch+1 in flight
    } else {
      WAIT_ASYNCCNT(0);
    }
    const float* xs  = xs0 + (ch & 1) * XCH;
    const float* wsm = ws0 + (ch & 1) * WCH;
    const int r0 = ch * RCH;
#pragma unroll
    for (int j = 0; j < RCH; j += 2) {
      {
        const float cw = cij[(r0 + j) * CC + c];
        v2f a0  = *(const v2f*)(xs + m * (RCH * II) + j * II + 2 * h);
        v2f a1  = *(const v2f*)(xs + m * (RCH * II) + j * II + 4 + 2 * h);
        v2f bv0 = *(const v2f*)(wsm + j * (OO * II) + m * II + 2 * h);
        v2f bv1 = *(const v2f*)(wsm + j * (OO * II) + m * II + 4 + 2 * h);
        bv0 *= cw;
        bv1 *= cw;
        acc0 = __builtin_amdgcn_wmma_f32_16x16x4_f32(false, a0, false, bv0,
                                                     (short)0, acc0, false, false);
        acc0 = __builtin_amdgcn_wmma_f32_16x16x4_f32(false, a1, false, bv1,
                                                     (short)0, acc0, false, false);
      }
      {
        const int j1 = j + 1;
        const float cw = cij[(r0 + j1) * CC + c];
        v2f a0  = *(const v2f*)(xs + m * (RCH * II) + j1 * II + 2 * h);
        v2f a1  = *(const v2f*)(xs + m * (RCH * II) + j1 * II + 4 + 2 * h);
        v2f bv0 = *(const v2f*)(wsm + j1 * (OO * II) + m * II + 2 * h);
        v2f bv1 = *(const v2f*)(wsm + j1 * (OO * II) + m * II + 4 + 2 * h);
        bv0 *= cw;
        bv1 *= cw;
        acc1 = __builtin_amdgcn_wmma_f32_16x16x4_f32(false, a0, false, bv0,
                                                     (short)0, acc1, false, false);
        acc1 = __builtin_amdgcn_wmma_f32_16x16x4_f32(false, a1, false, bv1,
                                                     (short)0, acc1, false, false);
      }
    }
  }
#else
  // Fallback: direct global loads (known-good path)
  const float* xrow = x + (size_t)(b0 + m) * RR * II;
  for (int r = 0; r < RR; r += 2) {
#pragma unroll
    for (int u = 0; u < 2; ++u) {
      const int rr = r + u;
      const float cw = cij[rr * CC + c];
      const float* xp = xrow + rr * II;
      v2f a0 = *(const v2f*)(xp + 2 * h);
      v2f a1 = *(const v2f*)(xp + 4 + 2 * h);
      const float* wp = W + ((size_t)rr * CC + c) * (OO * II) + m * II;
      v2f bv0 = *(const v2f*)(wp + 2 * h);
      v2f bv1 = *(const v2f*)(wp + 4 + 2 * h);
      bv0 *= cw;
      bv1 *= cw;
      if (u == 0) {
        acc0 = __builtin_amdgcn_wmma_f32_16x16x4_f32(false, a0, false, bv0,
                                                     (short)0, acc0, false, false);
        acc0 = __builtin_amdgcn_wmma_f32_16x16x4_f32(false, a1, false, bv1,
                                                     (short)0, acc0, false, false);
      } else {
        acc1 = __builtin_amdgcn_wmma_f32_16x16x4_f32(false, a0, false, bv0,
                                                     (short)0, acc1, false, false);
        acc1 = __builtin_amdgcn_wmma_f32_16x16x4_f32(false, a1, false, bv1,
                                                     (short)0, acc1, false, false);
      }
    }
  }
#endif

  v8f acc = acc0 + acc1;
  // D layout: VGPR t -> M = t (lanes 0-15) or t+8 (lanes 16-31), N = lane&15
#pragma unroll
  for (int t = 0; t < 8; ++t) {
    s[(size_t)(b0 + t + 8 * h) * (CC * OO) + c * OO + m] = acc[t];
  }
}

// ---------------------------------------------------------------------------
// v = squash(s), elementwise: v = s^2 * s / ((1+s^2) * sqrt(s^2))
// ---------------------------------------------------------------------------
__global__ void caps_squash_kernel(const float* __restrict__ s,
                                   float* __restrict__ dst, int n) {
  int i = blockIdx.x * blockDim.x + threadIdx.x;
  if (i < n) {
    float sv = s[i];
    float sq = sv * sv;
    dst[i] = sq * sv / ((1.0f + sq) * sqrtf(sq));
  }
}

// ---------------------------------------------------------------------------
// a[r,c] = sum_b sum_o u_hat[b,r,c,o] * v[b,c,o]   (mean 1/B applied later)
// u_hat tile recomputed on the fly with WMMA, dotted with v tile, wave-reduced.
// Each wave owns 4 routes -> no atomics, fully deterministic.
// ---------------------------------------------------------------------------
__global__ __launch_bounds__(128) void caps_a_kernel(
    const float* __restrict__ x, const float* __restrict__ W,
    const float* __restrict__ v, float* __restrict__ anew) {
  const int lane = threadIdx.x & 31;
  const int wid  = threadIdx.x >> 5;
  const int wgl  = blockIdx.x * 4 + wid;  // 0..287
  const int m = lane & 15;
  const int h = lane >> 4;
  const int r_begin = wgl * 4;

  for (int r = r_begin; r < r_begin + 4; ++r) {
    float partc[CC];
#pragma unroll
    for (int c = 0; c < CC; ++c) partc[c] = 0.0f;

    for (int b0 = 0; b0 < BB; b0 += 16) {
      const float* xp = x + ((size_t)(b0 + m) * RR + r) * II;
      v2f a0 = *(const v2f*)(xp + 2 * h);
      v2f a1 = *(const v2f*)(xp + 4 + 2 * h);
#pragma unroll
      for (int c = 0; c < CC; ++c) {
        const float* wp = W + ((size_t)r * CC + c) * (OO * II) + m * II;
        v2f bv0 = *(const v2f*)(wp + 2 * h);
        v2f bv1 = *(const v2f*)(wp + 4 + 2 * h);
        v8f acc = {};
        acc = __builtin_amdgcn_wmma_f32_16x16x4_f32(false, a0, false, bv0,
                                                    (short)0, acc, false, false);
        acc = __builtin_amdgcn_wmma_f32_16x16x4_f32(false, a1, false, bv1,
                                                    (short)0, acc, false, false);
        float t = 0.0f;
#pragma unroll
        for (int q = 0; q < 8; ++q) {
          t += acc[q] * v[(size_t)(b0 + q + 8 * h) * (CC * OO) + c * OO + m];
        }
        partc[c] += t;
      }
    }
#pragma unroll
    for (int c = 0; c < CC; ++c) {
      float t = partc[c];
#pragma unroll
      for (int off = 16; off > 0; off >>= 1) t += __shfl_xor(t, off, 32);
      if (lane == 0) anew[r * CC + c] = t;
    }
  }
}

// ---------------------------------------------------------------------------
// b_ij += a/B ; c_ij[:,c] = softmax over routes (axis r). One block per c.
// ---------------------------------------------------------------------------
__global__ __launch_bounds__(256) void caps_bsm_kernel(
    const float* __restrict__ anew, float* __restrict__ bij,
    float* __restrict__ cij) {
  __shared__ float red[256];
  const int c = blockIdx.x;
  const int tid = threadIdx.x;

  float mx = -3.4e38f;
  for (int r = tid; r < RR; r += 256) {
    float val = bij[r * CC + c] + anew[r * CC + c] * (1.0f / BB);
    bij[r * CC + c] = val;
    mx = fmaxf(mx, val);
  }
  red[tid] = mx;
  __syncthreads();
  for (int off = 128; off > 0; off >>= 1) {
    if (tid < off) red[tid] = fmaxf(red[tid], red[tid + off]);
    __syncthreads();
  }
  mx = red[0];
  __syncthreads();

  float sum = 0.0f;
  for (int r = tid; r < RR; r += 256) sum += __expf(bij[r * CC + c] - mx);
  red[tid] = sum;
  __syncthreads();
  for (int off = 128; off > 0; off >>= 1) {
    if (tid < off) red[tid] += red[tid + off];
    __syncthreads();
  }
  sum = red[0];

  const float inv = 1.0f / sum;
  for (int r = tid; r < RR; r += 256)
    cij[r * CC + c] = __expf(bij[r * CC + c] - mx) * inv;
}

__global__ void caps_init_kernel(float* __restrict__ bij,
                                 float* __restrict__ cij) {
  int i = blockIdx.x * blockDim.x + threadIdx.x;
  if (i < RR * CC) {
    bij[i] = 0.0f;
    cij[i] = 1.0f / (float)RR;
  }
}

extern "C" void kernel_launch(void* const* d_in, const int* in_sizes, int n_in,
                              void* d_out, int out_size, void* d_ws,
                              size_t ws_size, hipStream_t stream) {
  const float* x = (const float*)d_in[0];   // (B, R, I)
  const float* W = (const float*)d_in[1];   // (R, C, O, I)
  float* out = (float*)d_out;               // (B, C, O, 1) -> 40960 floats

  float* ws   = (float*)d_ws;
  float* s    = ws;                         // 40960
  float* v    = ws + 40960;                 // 40960
  float* bij  = ws + 81920;                 // 11520
  float* cij  = ws + 93440;                 // 11520
  float* anew = ws + 104960;                // 11520  (total ~466 KB)

  caps_init_kernel<<<(RR * CC + 255) / 256, 256, 0, stream>>>(bij, cij);

  for (int it = 0; it < 3; ++it) {
    caps_s_kernel<<<dim3(CC, 4), 128, 0, stream>>>(x, W, cij, s);
    float* dst = (it == 2) ? out : v;
    caps_squash_kernel<<<(BB * CC * OO + 255) / 256, 256, 0, stream>>>(
        s, dst, BB * CC * OO);
    if (it < 2) {
      caps_a_kernel<<<dim3(72), 128, 0, stream>>>(x, W, v, anew);
      caps_bsm_kernel<<<dim3(CC), 256, 0, stream>>>(anew, bij, cij);
    }
  }
}